// DGC_Attention_13597866459522
// MI455X (gfx1250) — compile-verified
//
#include <hip/hip_runtime.h>
#include <math.h>

// Problem constants (fixed by the reference).
static constexpr int kB = 16;    // batch
static constexpr int kC = 1024;  // channels
static constexpr int kS = 4096;  // h*w
static constexpr int kR = 64;    // bottleneck dim

typedef __attribute__((ext_vector_type(2))) float v2f;
typedef __attribute__((ext_vector_type(8))) float v8f;

// ---------------------------------------------------------------------------
// Kernel 1: cm[b][s] = sum_c x[b][c][s] * wm[c] + bm
// One thread per (b, s). Reads per wave are 32 contiguous floats (one 128B
// line) per c-iteration -> fully coalesced streaming of the 256MB tensor.
// ---------------------------------------------------------------------------
__global__ void __launch_bounds__(256) k_cm(const float* __restrict__ x,
                                            const float* __restrict__ wm,
                                            const float* __restrict__ bm,
                                            float* __restrict__ cm) {
  const int s = blockIdx.x * 256 + threadIdx.x;
  const int b = blockIdx.y;
  const float* xb = x + (size_t)b * kC * kS + s;
  float acc = 0.f;
#pragma unroll 4
  for (int c = 0; c < kC; ++c) {
    if ((c & 7) == 0 && c + 16 < kC) {
      __builtin_prefetch(xb + (size_t)(c + 16) * kS, 0, 0);
    }
    acc = fmaf(xb[(size_t)c * kS], wm[c], acc);
  }
  cm[b * kS + s] = acc + bm[0];
}

// ---------------------------------------------------------------------------
// Kernel 2: mask[b][s] = softmax_s(cm[b]) + 1/kS
// One 256-thread block per batch; LDS tree reductions for max and sum.
// ---------------------------------------------------------------------------
__global__ void __launch_bounds__(256) k_mask(const float* __restrict__ cm,
                                              float* __restrict__ mask) {
  const int b = blockIdx.x;
  const int tid = threadIdx.x;
  __shared__ float red[256];
  const float* row = cm + b * kS;
  float m = -1e30f;
  for (int i = tid; i < kS; i += 256) m = fmaxf(m, row[i]);
  red[tid] = m;
  __syncthreads();
  for (int w = 128; w > 0; w >>= 1) {
    if (tid < w) red[tid] = fmaxf(red[tid], red[tid + w]);
    __syncthreads();
  }
  const float mx = red[0];
  __syncthreads();
  float sum = 0.f;
  for (int i = tid; i < kS; i += 256) {
    float e = __expf(row[i] - mx);
    mask[b * kS + i] = e;
    sum += e;
  }
  red[tid] = sum;
  __syncthreads();
  for (int w = 128; w > 0; w >>= 1) {
    if (tid < w) red[tid] += red[tid + w];
    __syncthreads();
  }
  const float inv = 1.f / red[0];
  const float uni = 1.f / (float)kS;
  for (int i = tid; i < kS; i += 256) {
    mask[b * kS + i] = mask[b * kS + i] * inv + uni;
  }
}

// ---------------------------------------------------------------------------
// Kernel 3: context[b][c] = sum_s x[b][c][s] * mask[b][s]
// One 256-thread block per (b, c); coalesced 4096-element reduction.
// ---------------------------------------------------------------------------
__global__ void __launch_bounds__(256) k_context(const float* __restrict__ x,
                                                 const float* __restrict__ mask,
                                                 float* __restrict__ ctx) {
  const int c = blockIdx.x;
  const int b = blockIdx.y;
  const int tid = threadIdx.x;
  __shared__ float red[256];
  const float* xr = x + ((size_t)b * kC + c) * kS;
  const float* mk = mask + b * kS;
  float acc = 0.f;
  for (int i = tid; i < kS; i += 256) acc = fmaf(xr[i], mk[i], acc);
  red[tid] = acc;
  __syncthreads();
  for (int w = 128; w > 0; w >>= 1) {
    if (tid < w) red[tid] += red[tid + w];
    __syncthreads();
  }
  if (tid == 0) ctx[b * kC + c] = red[0];
}

// ---------------------------------------------------------------------------
// Kernel 4: MLP with f32 WMMA (V_WMMA_F32_16X16X4_F32).
//   t  = ctx(16x1024) @ w1^T(1024x64) + b1   -> LDS
//   t  = ReLU(LayerNorm_64(t) * g + b)
//   out= t(16x64) @ w2^T(64x1024) + b2
// One block of 128 threads = 4 full wave32s (EXEC all ones, as WMMA needs).
// Per-lane WMMA layouts (ISA 7.12.2, 32-bit):
//   A 16x4 : lane = m (0..15) | m (16..31); VGPR0 -> K = koff, VGPR1 -> K+1,
//            where koff = 0 for lanes 0-15, 2 for lanes 16-31.
//   B 4x16 : lane = n | n; same K striping as A.
//   C/D    : VGPR v holds row (v) for lanes 0-15, row (v+8) for lanes 16-31,
//            column = lane & 15.
// ---------------------------------------------------------------------------
__global__ void __launch_bounds__(128) k_mlp(const float* __restrict__ ctx,
                                             const float* __restrict__ w1,
                                             const float* __restrict__ b1,
                                             const float* __restrict__ ln_g,
                                             const float* __restrict__ ln_b,
                                             const float* __restrict__ w2,
                                             const float* __restrict__ b2,
                                             float* __restrict__ out) {
  __shared__ float t[kB * kR];  // 16 x 64 bottleneck activations
  const int tid = threadIdx.x;
  const int wave = tid >> 5;
  const int lane = tid & 31;
  const int half = (lane < 16) ? 0 : 1;
  const int mn = lane & 15;   // A-row (m) / B-col (n) for this lane
  const int koff = half * 2;  // K sub-offset for this half-wave

  // ---- Stage A: t = ctx @ w1^T + b1  (M=16, N=64 -> 4 waves x 16 cols) ----
  {
    const int n0 = wave * 16;
    v8f acc = {};
    for (int k = 0; k < kC; k += 4) {
      v2f a, bmat;
      a.x = ctx[mn * kC + k + koff];
      a.y = ctx[mn * kC + k + koff + 1];
      // B[k'][n] = w1^T[k'][n] = w1[n][k']
      bmat.x = w1[(n0 + mn) * kC + k + koff];
      bmat.y = w1[(n0 + mn) * kC + k + koff + 1];
      acc = __builtin_amdgcn_wmma_f32_16x16x4_f32(
          /*neg_a=*/false, a, /*neg_b=*/false, bmat,
          /*c_mod=*/(short)0, acc, /*reuse_a=*/false, /*reuse_b=*/false);
    }
#pragma unroll
    for (int v = 0; v < 8; ++v) {
      const int row = v + half * 8;
      const int col = n0 + mn;
      t[row * kR + col] = acc[v] + b1[col];
    }
  }
  __syncthreads();

  // ---- LayerNorm(64) * g + b, then ReLU; one thread per batch row ----
  if (tid < kB) {
    float mu = 0.f;
    for (int j = 0; j < kR; ++j) mu += t[tid * kR + j];
    mu *= (1.f / (float)kR);
    float var = 0.f;
    for (int j = 0; j < kR; ++j) {
      const float d = t[tid * kR + j] - mu;
      var += d * d;
    }
    var *= (1.f / (float)kR);
    const float inv = rsqrtf(var + 1e-5f);
    for (int j = 0; j < kR; ++j) {
      const float v = (t[tid * kR + j] - mu) * inv * ln_g[j] + ln_b[j];
      t[tid * kR + j] = fmaxf(v, 0.f);
    }
  }
  __syncthreads();

  // ---- Stage B: out = t @ w2^T + b2  (M=16, N=1024 -> 64 tiles / 4 waves) --
  for (int nt = wave; nt < kC / 16; nt += 4) {
    v8f acc = {};
    for (int k = 0; k < kR; k += 4) {
      v2f a, bmat;
      a.x = t[mn * kR + k + koff];
      a.y = t[mn * kR + k + koff + 1];
      // B[k'][n] = w2^T[k'][n] = w2[n][k']
      bmat.x = w2[(nt * 16 + mn) * kR + k + koff];
      bmat.y = w2[(nt * 16 + mn) * kR + k + koff + 1];
      acc = __builtin_amdgcn_wmma_f32_16x16x4_f32(
          false, a, false, bmat, (short)0, acc, false, false);
    }
#pragma unroll
    for (int v = 0; v < 8; ++v) {
      const int row = v + half * 8;
      const int col = nt * 16 + mn;
      out[row * kC + col] = acc[v] + b2[col];
    }
  }
}

// ---------------------------------------------------------------------------
// Launch
// ---------------------------------------------------------------------------
extern "C" void kernel_launch(void* const* d_in, const int* in_sizes, int n_in,
                              void* d_out, int out_size, void* d_ws, size_t ws_size,
                              hipStream_t stream) {
  const float* x    = (const float*)d_in[0];
  const float* wm   = (const float*)d_in[1];
  const float* bm   = (const float*)d_in[2];
  const float* w1   = (const float*)d_in[3];
  const float* b1   = (const float*)d_in[4];
  const float* ln_g = (const float*)d_in[5];
  const float* ln_b = (const float*)d_in[6];
  const float* w2   = (const float*)d_in[7];
  const float* b2   = (const float*)d_in[8];
  float* out = (float*)d_out;

  // Workspace layout: cm[16*4096] | mask[16*4096] | ctx[16*1024]  (~576 KB)
  float* cm   = (float*)d_ws;
  float* mask = cm + (size_t)kB * kS;
  float* ctx  = mask + (size_t)kB * kS;

  k_cm<<<dim3(kS / 256, kB), 256, 0, stream>>>(x, wm, bm, cm);
  k_mask<<<kB, 256, 0, stream>>>(cm, mask);
  k_context<<<dim3(kC, kB), 256, 0, stream>>>(x, mask, ctx);
  k_mlp<<<1, 128, 0, stream>>>(ctx, w1, b1, ln_g, ln_b, w2, b2, out);
}